// BayesianLayer_75333726371955
// MI455X (gfx1250) — compile-verified
//
#include <hip/hip_runtime.h>
#include <hip/hip_bf16.h>
#include <math.h>

#define B_SZ   32
#define R_SZ   2
#define IN_SZ  1024
#define OUT_SZ 1024

typedef __attribute__((ext_vector_type(2))) float v2f;
typedef __attribute__((ext_vector_type(4))) float v4f;
typedef __attribute__((ext_vector_type(8))) float v8f;

// ws float layout:
//  [0] = sum log(mix(w))        over eps tensor
//  [1] = sum eps^2              over eps tensor
//  [2] = sum log(sigma)         over (IN,OUT)
//  [3] = sum log(mix(bias))     over (B,R,OUT)
//  [4] = sum eps_bias^2         over (B,R,OUT)
//  [5] = sum log(sigma_bias)    over (OUT)
//  [64 ..] = sigma array (IN*OUT floats) if workspace is large enough

__device__ __forceinline__ float softplus_f(float v) {
    // log1p(exp(v)); ro ~ -8 so exp(v) is small, log1pf is accurate enough here
    return log1pf(__expf(v));
}

// ---------------------------------------------------------------------------
// Kernel 1: sigma = softplus(ro) (optionally stored), reduce sum(log sigma)
// ---------------------------------------------------------------------------
__global__ void sigma_stat_kernel(const float* __restrict__ ro,
                                  float* __restrict__ sigma_out,  // may be null
                                  float* __restrict__ acc,
                                  int write_sigma) {
    __shared__ float sred[256];
    int idx = blockIdx.x * 256 + threadIdx.x;   // IN*OUT = 1M, grid covers all
    float sg = softplus_f(ro[idx]);
    if (write_sigma) sigma_out[idx] = sg;
    sred[threadIdx.x] = __logf(sg);
    __syncthreads();
    for (int s = 128; s > 0; s >>= 1) {
        if (threadIdx.x < s) sred[threadIdx.x] += sred[threadIdx.x + s];
        __syncthreads();
    }
    if (threadIdx.x == 0) atomicAdd(&acc[2], sred[0]);
}

// ---------------------------------------------------------------------------
// Kernel 2: G[b,o] = sum_i x[b,i] * mu[i,o]  via V_WMMA_F32_16X16X4_F32
// M=32 (B), N=1024 (OUT), K=1024 (IN). One 16x16 tile per wave; 128 tiles.
// Launch: 32 blocks x 128 threads (4 waves/block).
// ---------------------------------------------------------------------------
__global__ void gemm_xmu_wmma(const float* __restrict__ x,
                              const float* __restrict__ mu,
                              float* __restrict__ out) {
    const int wave = threadIdx.x >> 5;
    const int lane = threadIdx.x & 31;
    const int lin  = blockIdx.x * 4 + wave;     // 0..127
    const int tm   = lin >> 6;                  // 0..1   (M tiles of 16)
    const int tn   = lin & 63;                  // 0..63  (N tiles of 16)
    const int m0   = tm << 4;
    const int n0   = tn << 4;
    const int half = lane >> 4;                 // 0/1
    const int l    = lane & 15;

    v8f c = {};
    for (int k = 0; k < IN_SZ; k += 4) {
        const int ka = k + 2 * half;
        v2f a, bm;
        // A (16x4, M x K): lanes 0-15 hold K={k,k+1}, lanes 16-31 hold K={k+2,k+3}
        a.x = x[(size_t)(m0 + l) * IN_SZ + ka];
        a.y = x[(size_t)(m0 + l) * IN_SZ + ka + 1];
        // B (4x16, K x N): row K striped across lanes within a VGPR
        bm.x = mu[(size_t)ka * OUT_SZ + n0 + l];
        bm.y = mu[(size_t)(ka + 1) * OUT_SZ + n0 + l];
        c = __builtin_amdgcn_wmma_f32_16x16x4_f32(
                /*neg_a=*/false, a, /*neg_b=*/false, bm,
                /*c_mod=*/(short)0, c, /*reuse_a=*/false, /*reuse_b=*/false);
    }
    // D layout: VGPR j holds row M = m0 + half*8 + j, col N = n0 + l
    #pragma unroll
    for (int j = 0; j < 8; ++j) {
        out[(size_t)(m0 + half * 8 + j) * OUT_SZ + n0 + l] = c[j];
    }
}

// ---------------------------------------------------------------------------
// Kernel 3: streaming pass over eps (256 MB, read once -> non-temporal loads
// so the stream does not evict the L2-resident sigma/mu arrays). Fuses:
//   out[b,o] += (1/R) * sum_{r,i} x[b,i] * eps * sigma      (atomic f32)
//   acc[0]   += sum log( exp(-w^2/2) + 0.5 exp(-w^2/8) ),  w = eps*sigma+mu
//   acc[1]   += sum eps^2
// Grid: (1, B, ISPLIT), block 256, 4 outputs (float4) per thread.
// ---------------------------------------------------------------------------
#define ISPLIT 16
__global__ void eps_stream_kernel(const float* __restrict__ x,
                                  const float* __restrict__ mu,
                                  const float* __restrict__ ro,
                                  const float* __restrict__ sigma,  // null -> inline softplus
                                  const float* __restrict__ eps,
                                  float* __restrict__ out,
                                  float* __restrict__ acc) {
    __shared__ float s_lm[256];
    __shared__ float s_q[256];

    const int tid = threadIdx.x;
    const int b   = blockIdx.y;
    const int o4  = tid * 4;                       // 256 threads * 4 = 1024 = OUT
    const int i0  = blockIdx.z * (IN_SZ / ISPLIT); // 64-row slab
    const int i1  = i0 + (IN_SZ / ISPLIT);

    v4f E = {0.f, 0.f, 0.f, 0.f};
    float lm = 0.f, q = 0.f;

    for (int r = 0; r < R_SZ; ++r) {
        const float* eb = eps + ((size_t)(b * R_SZ + r) * IN_SZ) * OUT_SZ;
        for (int i = i0; i < i1; ++i) {
            const size_t row = (size_t)i * OUT_SZ + o4;
            // eps: single-use 256MB stream -> non-temporal (TH=NT)
            const v4f e = __builtin_nontemporal_load((const v4f*)(eb + row));
            v4f sg;
            if (sigma) {
                sg = *(const v4f*)(sigma + row);
            } else {
                const v4f rv = *(const v4f*)(ro + row);
                sg.x = softplus_f(rv.x); sg.y = softplus_f(rv.y);
                sg.z = softplus_f(rv.z); sg.w = softplus_f(rv.w);
            }
            const v4f m = *(const v4f*)(mu + row);
            const float xb = x[(size_t)b * IN_SZ + i];

            float t, w, w2;
            t = e.x * sg.x; w = t + m.x; E.x = fmaf(xb, t, E.x);
            w2 = w * w;
            lm += __logf(__expf(-0.5f * w2) + 0.5f * __expf(-0.125f * w2));
            q = fmaf(e.x, e.x, q);

            t = e.y * sg.y; w = t + m.y; E.y = fmaf(xb, t, E.y);
            w2 = w * w;
            lm += __logf(__expf(-0.5f * w2) + 0.5f * __expf(-0.125f * w2));
            q = fmaf(e.y, e.y, q);

            t = e.z * sg.z; w = t + m.z; E.z = fmaf(xb, t, E.z);
            w2 = w * w;
            lm += __logf(__expf(-0.5f * w2) + 0.5f * __expf(-0.125f * w2));
            q = fmaf(e.z, e.z, q);

            t = e.w * sg.w; w = t + m.w; E.w = fmaf(xb, t, E.w);
            w2 = w * w;
            lm += __logf(__expf(-0.5f * w2) + 0.5f * __expf(-0.125f * w2));
            q = fmaf(e.w, e.w, q);
        }
    }

    const float invR = 1.0f / (float)R_SZ;
    float* ob = out + (size_t)b * OUT_SZ + o4;
    atomicAdd(ob + 0, E.x * invR);
    atomicAdd(ob + 1, E.y * invR);
    atomicAdd(ob + 2, E.z * invR);
    atomicAdd(ob + 3, E.w * invR);

    s_lm[tid] = lm; s_q[tid] = q;
    __syncthreads();
    for (int s = 128; s > 0; s >>= 1) {
        if (tid < s) { s_lm[tid] += s_lm[tid + s]; s_q[tid] += s_q[tid + s]; }
        __syncthreads();
    }
    if (tid == 0) { atomicAdd(&acc[0], s_lm[0]); atomicAdd(&acc[1], s_q[0]); }
}

// ---------------------------------------------------------------------------
// Kernel 4: bias path. idx over B*R*OUT = 65536.
//   bias = eps_bias*softplus(ro_bias) + mu_bias
//   out[b,o] += bias / R ; acc[3] += log-mix(bias) ; acc[4] += eps_bias^2
//   threads with idx<OUT also contribute acc[5] += log(sigma_bias[o])
// ---------------------------------------------------------------------------
__global__ void bias_kernel(const float* __restrict__ mu_bias,
                            const float* __restrict__ ro_bias,
                            const float* __restrict__ eps_bias,
                            float* __restrict__ out,
                            float* __restrict__ acc) {
    __shared__ float s_lm[256];
    __shared__ float s_q[256];
    const int tid = threadIdx.x;
    const int idx = blockIdx.x * 256 + tid;      // < B*R*OUT
    const int o   = idx & (OUT_SZ - 1);
    const int br  = idx >> 10;                   // b*R + r
    const int b   = br >> 1;

    const float sb = softplus_f(ro_bias[o]);
    const float e  = __builtin_nontemporal_load(eps_bias + idx);
    const float w  = fmaf(e, sb, mu_bias[o]);

    atomicAdd(&out[(size_t)b * OUT_SZ + o], w * (1.0f / (float)R_SZ));

    const float w2 = w * w;
    s_lm[tid] = __logf(__expf(-0.5f * w2) + 0.5f * __expf(-0.125f * w2));
    s_q[tid]  = e * e;
    if (idx < OUT_SZ) atomicAdd(&acc[5], __logf(sb));
    __syncthreads();
    for (int s = 128; s > 0; s >>= 1) {
        if (tid < s) { s_lm[tid] += s_lm[tid + s]; s_q[tid] += s_q[tid + s]; }
        __syncthreads();
    }
    if (tid == 0) { atomicAdd(&acc[3], s_lm[0]); atomicAdd(&acc[4], s_q[0]); }
}

// ---------------------------------------------------------------------------
// Kernel 5: finalize the two scalars.
// ---------------------------------------------------------------------------
__global__ void finalize_kernel(const float* __restrict__ acc,
                                float* __restrict__ out) {
    const float LOG_SQRT_2PI = 0.9189385332046727f;
    const float CMIX = -0.6931471805599453f - LOG_SQRT_2PI; // log(0.5/sqrt(2pi))
    const float Nw = (float)((size_t)B_SZ * R_SZ * IN_SZ * OUT_SZ); // 67108864
    const float Nb = (float)(B_SZ * R_SZ * OUT_SZ);                 // 65536
    const float denom = (float)(B_SZ * R_SZ);                       // 64

    const float mean_lm_w = acc[0] / Nw + CMIX;
    const float mean_lm_b = acc[3] / Nb + CMIX;
    const float log_prior = (mean_lm_w + mean_lm_b) / denom;

    const float mean_lp_w = -0.5f * acc[1] / Nw
                            - acc[2] / (float)(IN_SZ * OUT_SZ) - LOG_SQRT_2PI;
    const float mean_lp_b = -0.5f * acc[4] / Nb
                            - acc[5] / (float)OUT_SZ - LOG_SQRT_2PI;
    const float log_p = (mean_lp_w + mean_lp_b) / denom;

    out[(size_t)B_SZ * OUT_SZ + 0] = log_prior;
    out[(size_t)B_SZ * OUT_SZ + 1] = log_p;
}

// ---------------------------------------------------------------------------
extern "C" void kernel_launch(void* const* d_in, const int* in_sizes, int n_in,
                              void* d_out, int out_size, void* d_ws, size_t ws_size,
                              hipStream_t stream) {
    const float* x        = (const float*)d_in[0];   // (B, IN)
    const float* mu       = (const float*)d_in[1];   // (IN, OUT)
    const float* ro       = (const float*)d_in[2];   // (IN, OUT)
    const float* mu_bias  = (const float*)d_in[3];   // (1, OUT)
    const float* ro_bias  = (const float*)d_in[4];   // (1, OUT)
    const float* eps      = (const float*)d_in[5];   // (B, R, IN, OUT)
    const float* eps_bias = (const float*)d_in[6];   // (B, R, OUT)

    float* out = (float*)d_out;                      // 32*1024 + 2 floats
    float* acc = (float*)d_ws;

    const size_t sigma_bytes = (size_t)64 * 4 + (size_t)IN_SZ * OUT_SZ * 4;
    const int has_sigma = (ws_size >= sigma_bytes) ? 1 : 0;
    float* sigma_arr = has_sigma ? (acc + 64) : nullptr;

    // zero scalar accumulators
    hipMemsetAsync(d_ws, 0, 64 * sizeof(float), stream);

    // sigma precompute + sum(log sigma)
    sigma_stat_kernel<<<(IN_SZ * OUT_SZ) / 256, 256, 0, stream>>>(
        ro, sigma_arr, acc, has_sigma);

    // WMMA GEMM: out[b,o] = sum_i x[b,i]*mu[i,o]
    gemm_xmu_wmma<<<32, 128, 0, stream>>>(x, mu, out);

    // streaming eps pass
    eps_stream_kernel<<<dim3(1, B_SZ, ISPLIT), 256, 0, stream>>>(
        x, mu, ro, sigma_arr, eps, out, acc);

    // bias path
    bias_kernel<<<(B_SZ * R_SZ * OUT_SZ) / 256, 256, 0, stream>>>(
        mu_bias, ro_bias, eps_bias, out, acc);

    // scalars
    finalize_kernel<<<1, 1, 0, stream>>>(acc, out);
}